// MentionRankingModel_59536836657581
// MI455X (gfx1250) — compile-verified
//
#include <hip/hip_runtime.h>
#include <hip/hip_bf16.h>

// ---------------------------------------------------------------------------
// MentionRanking loss, N=4096.  Bandwidth-bound: ~100MB essential traffic for
// the row pass (+34MB for the global min) -> ~5.7us floor at 23.3 TB/s.
// No matmul structure => WMMA inapplicable; the CDNA5 path exercised is the
// Tensor Data Mover: per-row DMA of the 16KB mask row AND the ana-row slice
// into LDS (tensor_load_to_lds x2 + s_wait_tensorcnt), with wave32 shuffle
// reductions for the two row-max phases.
// ---------------------------------------------------------------------------

#define NN 4096
#define TRI_M (NN * (NN - 1) / 2) /* 8386560 */
#define NBLK_MIN 512
#define MIN_STRIDE (NBLK_MIN * 256) /* compile-time grid stride */

// ---- CDNA5 TDM availability / arity discrimination ------------------------
#if defined(__has_builtin)
#if __has_builtin(__builtin_amdgcn_tensor_load_to_lds) && \
    __has_builtin(__builtin_amdgcn_s_wait_tensorcnt)
#define HAVE_TDM 1
#if __has_include(<hip/amd_detail/amd_gfx1250_TDM.h>)
#define TDM_ARGS 6 /* amdgpu-toolchain (clang-23 / therock-10.0 headers) */
#else
#define TDM_ARGS 5 /* ROCm 7.2 (clang-22) */
#endif
#endif
#endif

typedef unsigned int u32x4 __attribute__((ext_vector_type(4)));
typedef int i32x4 __attribute__((ext_vector_type(4)));
typedef int i32x8 __attribute__((ext_vector_type(8)));

// ---- wave32 reductions -----------------------------------------------------
__device__ __forceinline__ float wave_max32(float v) {
#pragma unroll
  for (int off = 16; off > 0; off >>= 1) v = fmaxf(v, __shfl_xor(v, off, 32));
  return v;
}
__device__ __forceinline__ float wave_min32(float v) {
#pragma unroll
  for (int off = 16; off > 0; off >>= 1) v = fminf(v, __shfl_xor(v, off, 32));
  return v;
}

#ifdef HAVE_TDM
// Issue a 1-D TDM copy: nelem f32 from `gsrc` (global) to `ldst` (LDS).
// Caller must be a full wave; completion via s_wait_tensorcnt.
__device__ __forceinline__ void tdm_copy_row_f32(const float* gsrc, float* ldst,
                                                 unsigned nelem) {
  const unsigned lds_addr = (unsigned)(unsigned long long)(const void*)ldst;
  const unsigned long long ga = (unsigned long long)(const void*)gsrc;
  u32x4 g0;
  g0.x = 1u;                                  // count=1, user desc, no gather
  g0.y = lds_addr;                            // D#.lds_addr (bytes)
  g0.z = (unsigned)(ga & 0xFFFFFFFFu);        // global_addr[31:0]
  g0.w = (unsigned)((ga >> 32) & 0x01FFFFFFu) // global_addr[56:32]
         | (2u << 30);                        // type = 2 ("image")
  i32x8 g1;
  g1[0] = (2 << 16);                     // data_size = 4B; no multicast/pad
  g1[1] = (int)((nelem & 0xFFFFu) << 16);      // tensor_dim0[15:0]
  g1[2] = (int)((nelem >> 16) | (1u << 16));   // tensor_dim0 hi | tensor_dim1=1
  g1[3] = (int)((nelem & 0xFFFFu) << 16);      // tile_dim0 = nelem
  g1[4] = 0;                                   // tile_dim1/2 = 0 (unused)
  g1[5] = (int)nelem;                          // tensor_dim0_stride
  g1[6] = 0;
  g1[7] = 0;
  i32x4 z4 = {0, 0, 0, 0};
#if TDM_ARGS == 6
  i32x8 z8 = {0, 0, 0, 0, 0, 0, 0, 0};
  __builtin_amdgcn_tensor_load_to_lds(g0, g1, z4, z4, z8, 0);
#else
  __builtin_amdgcn_tensor_load_to_lds(g0, g1, z4, z4, 0);
#endif
}
#endif

// ---------------------------------------------------------------------------
// Pass 1a: block-partial min over {ana, eps, 0}.  The upper triangle of the
// score matrix is zero and participates in scores.min(), hence the 0.0f seed.
// Fixed compile-time stride avoids the runtime trip-count division.
// ---------------------------------------------------------------------------
__global__ __launch_bounds__(256) void kmin_partial(
    const float* __restrict__ ana, const float* __restrict__ eps,
    float* __restrict__ partials) {
  __shared__ float red[8];
  float mn = 0.0f;
  const int tid = blockIdx.x * 256 + threadIdx.x;
  const float4* a4 = (const float4*)ana; // TRI_M divisible by 4
  for (int k = tid; k < TRI_M / 4; k += MIN_STRIDE) {
    float4 v = a4[k];
    mn = fminf(mn, fminf(fminf(v.x, v.y), fminf(v.z, v.w)));
  }
  if (tid < NN) mn = fminf(mn, eps[tid]); // NN < MIN_STRIDE: single strip
  mn = wave_min32(mn);
  if ((threadIdx.x & 31) == 0) red[threadIdx.x >> 5] = mn;
  __syncthreads();
  if (threadIdx.x == 0) {
    float r = red[0];
#pragma unroll
    for (int k = 1; k < 8; ++k) r = fminf(r, red[k]);
    partials[blockIdx.x] = r;
  }
}

// Pass 1b: reduce NBLK_MIN partials -> ws[0]
__global__ __launch_bounds__(256) void kmin_final(
    const float* __restrict__ partials, float* __restrict__ minv) {
  __shared__ float red[8];
  float mn = 0.0f;
  for (int k = threadIdx.x; k < NBLK_MIN; k += 256) mn = fminf(mn, partials[k]);
  mn = wave_min32(mn);
  if ((threadIdx.x & 31) == 0) red[threadIdx.x >> 5] = mn;
  __syncthreads();
  if (threadIdx.x == 0) {
    float r = red[0];
#pragma unroll
    for (int k = 1; k < 8; ++k) r = fminf(r, red[k]);
    minv[0] = r;
  }
}

// ---------------------------------------------------------------------------
// Pass 2: one 256-thread block (8 wave32) per row i.
//   - wave 0 issues two TDM descriptors: mask row (16KB) -> s_mask,
//     ana[tri(i) .. tri(i)+i) -> s_score[0..i)
//   - remaining threads fill s_score[i]=eps[i], s_score[j>i]=0
//   - s_wait_tensorcnt 0; barrier
//   - phase 1: best_correct = max_j( m*score + (1-m)*minScore )
//   - phase 2: loss        = max_j( (1-m)*pc * (1 + score - best_correct) )
// ---------------------------------------------------------------------------
__global__ __launch_bounds__(256) void krow(
    const float* __restrict__ eps, const float* __restrict__ ana,
    const float* __restrict__ mask, const float* __restrict__ lc,
    const float* __restrict__ fnc, const float* __restrict__ minv,
    float* __restrict__ out) {
  __shared__ float s_mask[NN];
  __shared__ float s_score[NN];
  __shared__ float s_red[8];

  const int i = blockIdx.x;
  const int t = threadIdx.x;
  const float* mrow = mask + (size_t)i * NN;
  const size_t triBase = ((size_t)i * (size_t)(i - 1)) >> 1; // i=0 -> 0

#ifdef HAVE_TDM
  if (t < 32) { // wave 0 drives the DMA engine
    tdm_copy_row_f32(mrow, s_mask, NN);
    if (i > 0) tdm_copy_row_f32(ana + triBase, s_score, (unsigned)i);
  }
  // Tail of the score row (TDM owns j < i): j == i -> eps, j > i -> 0.
  for (int j = t; j < NN; j += 256) {
    if (j > i)       s_score[j] = 0.0f;
    else if (j == i) s_score[j] = eps[i];
  }
  if (t < 32) __builtin_amdgcn_s_wait_tensorcnt(0);
#else
  // Fallback: cooperative b128 copy of the mask row + coalesced ana loads.
  for (int k = t; k < NN / 4; k += 256)
    ((float4*)s_mask)[k] = ((const float4*)mrow)[k];
  if ((size_t)t * 16 < (size_t)i)
    __builtin_prefetch(&ana[triBase + (size_t)t * 16], 0, 1);
  for (int j = t; j < NN; j += 256) {
    float sc = 0.0f;
    if (j < i)       sc = ana[triBase + j];
    else if (j == i) sc = eps[i];
    s_score[j] = sc;
  }
#endif
  __syncthreads();

  // Phase 1: best_correct
  const float mnS = minv[0];
  float best = -__builtin_inff();
  for (int j = t; j < NN; j += 256) {
    float m = s_mask[j];
    best = fmaxf(best, m * s_score[j] + (1.0f - m) * mnS);
  }
  best = wave_max32(best);
  const int wid = t >> 5;
  if ((t & 31) == 0) s_red[wid] = best;
  __syncthreads();
  float bc = s_red[0];
#pragma unroll
  for (int k = 1; k < 8; ++k) bc = fmaxf(bc, s_red[k]);
  __syncthreads(); // protect s_red before reuse

  // Phase 2: hinge-style loss row max
  const float mdiag = s_mask[i];
  const float rc = mdiag * lc[0] + (1.0f - mdiag) * lc[1];
  const float fn = fnc[0];
  float lmax = -__builtin_inff();
  for (int j = t; j < NN; j += 256) {
    float m = s_mask[j];
    float pc = (j < i) ? rc : ((j == i) ? fn : 0.0f);
    float cost = (1.0f - m) * pc;
    lmax = fmaxf(lmax, cost * (1.0f + s_score[j] - bc));
  }
  lmax = wave_max32(lmax);
  if ((t & 31) == 0) s_red[wid] = lmax;
  __syncthreads();
  if (t == 0) {
    float r = s_red[0];
#pragma unroll
    for (int k = 1; k < 8; ++k) r = fmaxf(r, s_red[k]);
    out[i] = r;
  }
}

// ---------------------------------------------------------------------------
extern "C" void kernel_launch(void* const* d_in, const int* in_sizes, int n_in,
                              void* d_out, int out_size, void* d_ws,
                              size_t ws_size, hipStream_t stream) {
  const float* eps  = (const float*)d_in[0]; // [4096]
  const float* ana  = (const float*)d_in[1]; // [8386560]
  const float* mask = (const float*)d_in[2]; // [4096*4096]
  const float* lc   = (const float*)d_in[3]; // [2]
  const float* fnc  = (const float*)d_in[4]; // [1]
  float* out = (float*)d_out;                // [4096]

  float* w = (float*)d_ws;
  float* minv = w;           // w[0]: global min of score matrix
  float* partials = w + 64;  // NBLK_MIN block partials

  kmin_partial<<<NBLK_MIN, 256, 0, stream>>>(ana, eps, partials);
  kmin_final<<<1, 256, 0, stream>>>(partials, minv);
  krow<<<NN, 256, 0, stream>>>(eps, ana, mask, lc, fnc, minv, out);
}